// YOLOXHat_11931419148639
// MI455X (gfx1250) — compile-verified
//
#include <hip/hip_runtime.h>
#include <hip/hip_bf16.h>
#include <stdint.h>

#define BATCH 32
#define NPRI  8400
#define NCLS  80
#define KTOP  1000
#define KPAD  1008      // 63 tiles of 16
#define NSORT 16384
#define CONF_THR 0.65f
#define IOU_THRV 0.65f

typedef float v2f __attribute__((ext_vector_type(2)));
typedef float v8f __attribute__((ext_vector_type(8)));
typedef int   v4i __attribute__((vector_size(16)));

__device__ __forceinline__ float sigm(float x) { return 1.0f / (1.0f + expf(-x)); }

// ---------------------------------------------------------------------------
// CDNA5 async global->LDS helpers (GLOBAL_LOAD_ASYNC_TO_LDS_*, ASYNCcnt).
// Signatures per hipcc diagnostic: (T __device__* gsrc, T __shared__* ldst,
// imm offset, imm cpol) with T = int (b32) / int4 (b128).
// ---------------------------------------------------------------------------
#if defined(__has_builtin)
#if __has_builtin(__builtin_amdgcn_global_load_async_to_lds_b128) && \
    __has_builtin(__builtin_amdgcn_global_load_async_to_lds_b32)
#define HAVE_ASYNC_LDS 1
#endif
#endif

__device__ __forceinline__ void lds_copy_b128(void* lds, const void* g) {
#ifdef HAVE_ASYNC_LDS
  __builtin_amdgcn_global_load_async_to_lds_b128(
      (__attribute__((address_space(1))) v4i*)g,
      (__attribute__((address_space(3))) v4i*)lds, 0, 0);
#else
  *(float4*)lds = *(const float4*)g;
#endif
}

__device__ __forceinline__ void lds_copy_b32(void* lds, const void* g) {
#ifdef HAVE_ASYNC_LDS
  __builtin_amdgcn_global_load_async_to_lds_b32(
      (__attribute__((address_space(1))) int*)g,
      (__attribute__((address_space(3))) int*)lds, 0, 0);
#else
  *(float*)lds = *(const float*)g;
#endif
}

__device__ __forceinline__ void lds_copy_wait() {
#ifdef HAVE_ASYNC_LDS
#if __has_builtin(__builtin_amdgcn_s_wait_asynccnt)
  __builtin_amdgcn_s_wait_asynccnt(0);
#else
  asm volatile("s_wait_asynccnt 0" ::: "memory");
#endif
#endif
}

// ---------------------------------------------------------------------------
// Kernel 1: per-(image,prior): class max/argmax, obj, score, decode box.
// ---------------------------------------------------------------------------
__global__ void k_pre(const float* __restrict__ cls0, const float* __restrict__ cls1,
                      const float* __restrict__ cls2, const float* __restrict__ bb0,
                      const float* __restrict__ bb1, const float* __restrict__ bb2,
                      const float* __restrict__ ob0, const float* __restrict__ ob1,
                      const float* __restrict__ ob2,
                      float* __restrict__ sc, int* __restrict__ lab,
                      float* __restrict__ boxes) {
  int t = blockIdx.x * blockDim.x + threadIdx.x;
  if (t >= BATCH * NPRI) return;
  int b = t / NPRI, n = t - b * NPRI;
  const float *cp, *bp, *op;
  int HW, W, hw; float s;
  if (n < 6400)      { HW = 6400; W = 80; s = 8.f;  hw = n;        cp = cls0; bp = bb0; op = ob0; }
  else if (n < 8000) { HW = 1600; W = 40; s = 16.f; hw = n - 6400; cp = cls1; bp = bb1; op = ob1; }
  else               { HW = 400;  W = 20; s = 32.f; hw = n - 8000; cp = cls2; bp = bb2; op = ob2; }
  int h = hw / W, w = hw - h * W;

  const float* c = cp + (size_t)b * NCLS * HW + hw;
  const float* bpp = bp + (size_t)b * 4 * HW + hw;
  __builtin_prefetch(bpp, 0, 0);              // global_prefetch_b8 for the box row

  float m = c[0]; int ml = 0;
  #pragma unroll 4
  for (int cc = 1; cc < NCLS; ++cc) {
    float v = c[(size_t)cc * HW];
    if (v > m) { m = v; ml = cc; }
  }
  float ol = op[(size_t)b * HW + hw];
  float score = sigm(m) * sigm(ol);
  float scv = (score >= CONF_THR) ? score : 0.0f;   // jnp.where(valid, score, 0)

  float bx = bpp[0], by = bpp[(size_t)HW], bw = bpp[(size_t)2 * HW], bh = bpp[(size_t)3 * HW];
  float cx = (float)w * s + bx * s;
  float cy = (float)h * s + by * s;
  float hw2 = expf(bw) * s * 0.5f;
  float hh2 = expf(bh) * s * 0.5f;
  float* o = boxes + (size_t)t * 4;
  o[0] = cx - hw2; o[1] = cy - hh2; o[2] = cx + hw2; o[3] = cy + hh2;
  sc[t] = scv; lab[t] = ml;
}

// ---------------------------------------------------------------------------
// Kernel 2: build sort keys  (score_bits << 32) | (~idx)  -> exact top_k order
// ---------------------------------------------------------------------------
__global__ void k_keys(const float* __restrict__ sc,
                       unsigned long long* __restrict__ keys) {
  int t = blockIdx.x * blockDim.x + threadIdx.x;
  if (t >= BATCH * NSORT) return;
  int b = t >> 14, n = t & (NSORT - 1);
  unsigned long long k = 0ull;
  if (n < NPRI) {
    unsigned sb = __float_as_uint(sc[(size_t)b * NPRI + n]);  // scores >= 0: bits monotone
    k = ((unsigned long long)sb << 32) | (unsigned long long)(0xFFFFFFFFu - (unsigned)n);
  }
  keys[t] = k;
}

// ---------------------------------------------------------------------------
// Kernel 3: per-image bitonic sort (descending) of 16384 u64 keys in ws.
// ---------------------------------------------------------------------------
__global__ void k_sort(unsigned long long* __restrict__ keys) {
  unsigned long long* a = keys + (size_t)blockIdx.x * NSORT;
  for (int k = 2; k <= NSORT; k <<= 1) {
    for (int j = k >> 1; j > 0; j >>= 1) {
      for (int i = threadIdx.x; i < NSORT; i += blockDim.x) {
        int l = i ^ j;
        if (l > i) {
          unsigned long long x = a[i], y = a[l];
          bool desc = ((i & k) == 0);
          if ((x < y) == desc) { a[i] = y; a[l] = x; }
        }
      }
      __syncthreads();
    }
  }
}

// ---------------------------------------------------------------------------
// Kernel 4: gather top-KPAD per image, max|coord|+1 reduction, offset boxes,
// areas. One block (256 thr) per image.
// ---------------------------------------------------------------------------
__global__ void k_gather(const unsigned long long* __restrict__ keys,
                         const float* __restrict__ boxes, const int* __restrict__ lab,
                         float* __restrict__ tb, float* __restrict__ tsc,
                         int* __restrict__ tlab, float* __restrict__ obx,
                         float* __restrict__ area) {
  int b = blockIdx.x, tid = threadIdx.x;
  __shared__ float red[256];
  float mloc = 0.f;
  for (int k = tid; k < KPAD; k += 256) {
    float x0 = 0.f, y0 = 0.f, x1 = 0.f, y1 = 0.f, s = 0.f; int l = 0;
    if (k < KTOP) {
      unsigned long long key = keys[(size_t)b * NSORT + k];
      unsigned n = 0xFFFFFFFFu - (unsigned)key;
      s = __uint_as_float((unsigned)(key >> 32));
      const float* p = boxes + ((size_t)b * NPRI + n) * 4;
      x0 = p[0]; y0 = p[1]; x1 = p[2]; y1 = p[3];
      l = lab[(size_t)b * NPRI + n];
      mloc = fmaxf(mloc, fmaxf(fmaxf(fabsf(x0), fabsf(y0)), fmaxf(fabsf(x1), fabsf(y1))));
    }
    size_t o = (size_t)b * KPAD + k;
    tb[o * 4 + 0] = x0; tb[o * 4 + 1] = y0; tb[o * 4 + 2] = x1; tb[o * 4 + 3] = y1;
    tsc[o] = s; tlab[o] = l;
  }
  red[tid] = mloc; __syncthreads();
  for (int st = 128; st > 0; st >>= 1) {
    if (tid < st) red[tid] = fmaxf(red[tid], red[tid + st]);
    __syncthreads();
  }
  float maxc = red[0] + 1.0f;
  for (int k = tid; k < KPAD; k += 256) {
    size_t o = (size_t)b * KPAD + k;
    float off = (k < KTOP) ? (float)tlab[o] * maxc : 0.f;
    float x0 = tb[o * 4 + 0] + off, y0 = tb[o * 4 + 1] + off;
    float x1 = tb[o * 4 + 2] + off, y1 = tb[o * 4 + 3] + off;
    obx[o * 4 + 0] = x0; obx[o * 4 + 1] = y0; obx[o * 4 + 2] = x1; obx[o * 4 + 3] = y1;
    area[o] = (x1 - x0) * (y1 - y0);
  }
}

// ---------------------------------------------------------------------------
// Kernel 5: suppression bitmask via WMMA + async LDS staging.
// Block = 8 waves sharing one 16-row ti and a 256-wide j-range. The block
// async-copies its j-boxes/areas + i-boxes/areas into LDS; each wave then
// runs two V_WMMA_F32_16X16X4_F32 for sums[i][j] = area_i + area_j and packs
// (inter > thr*(union+eps)) bits with ballot into u32 suppression words.
// ---------------------------------------------------------------------------
__global__ void k_iou(const float* __restrict__ obx, const float* __restrict__ area,
                      unsigned int* __restrict__ sup) {
  __shared__ __align__(16) float s_jb[256 * 4];  // j boxes for block's 256-wide range
  __shared__ float s_ja[256];                    // j areas
  __shared__ __align__(16) float s_ib[16 * 4];   // i boxes (one ti)
  __shared__ float s_ia[16];                     // i areas

  int b = blockIdx.y;
  int tid = threadIdx.x;
  int wv0 = blockIdx.x * 8;            // 8 waves/block, 8 | 32 -> same ti in block
  int ti = wv0 >> 5;                   // 0..62
  int tjp0 = wv0 & 31;                 // multiple of 8
  int j0blk = tjp0 * 32;               // block j base (256 wide)
  int i0 = ti << 4;
  const float* obb = obx + (size_t)b * KPAD * 4;
  const float* ab  = area + (size_t)b * KPAD;

  // ---- cooperative async stage: 256 lanes x b128 + b32, plus i-tile ----
  int jg = j0blk + tid; if (jg >= KPAD) jg = KPAD - 1;   // clamp (pads unused)
  lds_copy_b128(&s_jb[tid * 4], obb + (size_t)jg * 4);
  lds_copy_b32(&s_ja[tid], ab + jg);
  if (tid < 16) {
    lds_copy_b128(&s_ib[tid * 4], obb + (size_t)(i0 + tid) * 4);
    lds_copy_b32(&s_ia[tid], ab + (i0 + tid));
  }
  lds_copy_wait();      // s_wait_asynccnt 0 (per wave)
  __syncthreads();

  int wlocal = tid >> 5;               // wave in block 0..7
  int tjp = tjp0 + wlocal;             // word column 0..31
  int lane = tid & 31;
  bool hi = lane >= 16;
  int jloc = lane & 15;

  // Preload this lane's 8 row-boxes (C/D layout: M = r + 8*(lane>=16), N = lane%16)
  float ix0[8], iy0[8], ix1[8], iy1[8];
  #pragma unroll
  for (int r = 0; r < 8; ++r) {
    int ir = r + (hi ? 8 : 0);
    ix0[r] = s_ib[ir * 4 + 0]; iy0[r] = s_ib[ir * 4 + 1];
    ix1[r] = s_ib[ir * 4 + 2]; iy1[r] = s_ib[ir * 4 + 3];
  }

  // A (16x4 f32): lanes 0-15 hold M=lane, VGPR0=K0 (area_i), VGPR1=K1 (1.0)
  v2f A; A.x = hi ? 0.f : s_ia[jloc]; A.y = hi ? 0.f : 1.0f;

  unsigned bal[2][8];
  #pragma unroll
  for (int half = 0; half < 2; ++half) {
    int tj = 2 * tjp + half;
    if (tj < 63) {
      int jl = wlocal * 32 + half * 16 + jloc;   // LDS-local j index
      // B (4x16 f32): lanes 0-15 hold K0 (1.0) and K1 (area_j)
      v2f Bv; Bv.x = hi ? 0.f : 1.0f; Bv.y = hi ? 0.f : s_ja[jl];
      v8f C = {0.f, 0.f, 0.f, 0.f, 0.f, 0.f, 0.f, 0.f};
      v8f S = __builtin_amdgcn_wmma_f32_16x16x4_f32(
          /*neg_a=*/false, A, /*neg_b=*/false, Bv,
          /*c_mod=*/(short)0, C, /*reuse_a=*/false, /*reuse_b=*/false);

      float jx0 = s_jb[jl * 4 + 0], jy0 = s_jb[jl * 4 + 1];
      float jx1 = s_jb[jl * 4 + 2], jy1 = s_jb[jl * 4 + 3];
      #pragma unroll
      for (int r = 0; r < 8; ++r) {
        float tlx = fmaxf(ix0[r], jx0), tly = fmaxf(iy0[r], jy0);
        float brx = fminf(ix1[r], jx1), bry = fminf(iy1[r], jy1);
        float iw = fmaxf(brx - tlx, 0.f), ih = fmaxf(bry - tly, 0.f);
        float inter = iw * ih;
        float uni = S[r] - inter;                 // area_i + area_j - inter (WMMA sum)
        // iou > thr  <=>  inter > thr*(union+eps)   (denominator always > 0)
        bal[half][r] = __builtin_amdgcn_ballot_w32(inter > IOU_THRV * (uni + 1e-6f));
      }
    } else {
      #pragma unroll
      for (int r = 0; r < 8; ++r) bal[half][r] = 0u;
    }
  }

  if (lane == 0) {
    unsigned int* srow = sup + (size_t)b * KPAD * 32;
    #pragma unroll
    for (int r = 0; r < 8; ++r) {
      // ballot bits[15:0] = rows i0+r (lanes 0-15); bits[31:16] = rows i0+r+8
      unsigned lo = (bal[0][r] & 0xFFFFu) | (bal[1][r] << 16);
      unsigned hw = (bal[0][r] >> 16) | (bal[1][r] & 0xFFFF0000u);
      srow[(size_t)(i0 + r) * 32 + tjp] = lo;
      srow[(size_t)(i0 + r + 8) * 32 + tjp] = hw;
    }
  }
}

// ---------------------------------------------------------------------------
// Kernel 6: greedy sequential suppression scan + output. One wave per image;
// lane w owns keep word w. ds_bpermute broadcasts the keep/valid bit of row i.
// ---------------------------------------------------------------------------
__global__ void k_nms(const unsigned int* __restrict__ sup, const float* __restrict__ tsc,
                      const float* __restrict__ tb, const int* __restrict__ tlab,
                      float* __restrict__ out) {
  int b = blockIdx.x;
  int lane = threadIdx.x;  // blockDim == 32
  const unsigned int* S = sup + (size_t)b * KPAD * 32;
  const float* ts = tsc + (size_t)b * KPAD;

  unsigned keepw = 0xFFFFFFFFu;
  unsigned vw = 0u;
  #pragma unroll
  for (int p = 0; p < 32; ++p) {
    int k = lane * 32 + p;
    if (k < KTOP && ts[k] >= CONF_THR) vw |= (1u << p);
  }

  for (int i = 0; i < KTOP; ++i) {
    if (i + 1 < KTOP)
      __builtin_prefetch(&S[(size_t)(i + 1) * 32 + lane], 0, 0);  // hide row latency
    int wi = i >> 5, bi = i & 31;
    unsigned kv = (unsigned)__builtin_amdgcn_ds_bpermute(wi << 2, (int)keepw);
    unsigned vv = (unsigned)__builtin_amdgcn_ds_bpermute(wi << 2, (int)vw);
    bool act = ((kv >> bi) & 1u) && ((vv >> bi) & 1u);
    if (act) {
      unsigned s = S[(size_t)i * 32 + lane];
      if (lane < wi) s = 0u;
      else if (lane == wi) s = (bi == 31) ? 0u : (s & (0xFFFFFFFFu << (bi + 1)));  // j > i only
      keepw &= ~s;
    }
  }

  float* dets = out;                               // [32][1000][5]
  float* labsO = out + (size_t)BATCH * KTOP * 5;   // [32][1000]
  float* keepO = out + (size_t)BATCH * KTOP * 6;   // [32][1000]
  for (int k = lane; k < KTOP; k += 32) {
    int wk = k >> 5, bk = k & 31;
    unsigned kw2 = (unsigned)__builtin_amdgcn_ds_bpermute(wk << 2, (int)keepw);
    unsigned vw2 = (unsigned)__builtin_amdgcn_ds_bpermute(wk << 2, (int)vw);
    bool fin = ((kw2 >> bk) & 1u) && ((vw2 >> bk) & 1u);
    size_t o = (size_t)b * KPAD + k;
    size_t od = (size_t)b * KTOP + k;
    if (fin) {
      dets[od * 5 + 0] = tb[o * 4 + 0];
      dets[od * 5 + 1] = tb[o * 4 + 1];
      dets[od * 5 + 2] = tb[o * 4 + 2];
      dets[od * 5 + 3] = tb[o * 4 + 3];
      dets[od * 5 + 4] = ts[k];
      labsO[od] = (float)tlab[o];
      keepO[od] = 1.0f;
    } else {
      dets[od * 5 + 0] = 0.f; dets[od * 5 + 1] = 0.f; dets[od * 5 + 2] = 0.f;
      dets[od * 5 + 3] = 0.f; dets[od * 5 + 4] = 0.f;
      labsO[od] = -1.0f;
      keepO[od] = 0.0f;
    }
  }
}

// ---------------------------------------------------------------------------
extern "C" void kernel_launch(void* const* d_in, const int* in_sizes, int n_in,
                              void* d_out, int out_size, void* d_ws, size_t ws_size,
                              hipStream_t stream) {
  const float* cls0 = (const float*)d_in[0];
  const float* cls1 = (const float*)d_in[1];
  const float* cls2 = (const float*)d_in[2];
  const float* bb0  = (const float*)d_in[3];
  const float* bb1  = (const float*)d_in[4];
  const float* bb2  = (const float*)d_in[5];
  const float* ob0  = (const float*)d_in[6];
  const float* ob1  = (const float*)d_in[7];
  const float* ob2  = (const float*)d_in[8];

  char* ws = (char*)d_ws;
  size_t off = 0;
  float* sc = (float*)(ws + off);                 off += (size_t)BATCH * NPRI * 4;
  int* lab = (int*)(ws + off);                    off += (size_t)BATCH * NPRI * 4;
  float* boxes = (float*)(ws + off);              off += (size_t)BATCH * NPRI * 16;
  unsigned long long* keys =
      (unsigned long long*)(ws + off);            off += (size_t)BATCH * NSORT * 8;
  float* tb = (float*)(ws + off);                 off += (size_t)BATCH * KPAD * 16;
  float* tsc = (float*)(ws + off);                off += (size_t)BATCH * KPAD * 4;
  int* tlab = (int*)(ws + off);                   off += (size_t)BATCH * KPAD * 4;
  float* obx = (float*)(ws + off);                off += (size_t)BATCH * KPAD * 16;
  float* area = (float*)(ws + off);               off += (size_t)BATCH * KPAD * 4;
  unsigned int* sup = (unsigned int*)(ws + off);  off += (size_t)BATCH * KPAD * 32 * 4;

  k_pre<<<(BATCH * NPRI + 255) / 256, 256, 0, stream>>>(
      cls0, cls1, cls2, bb0, bb1, bb2, ob0, ob1, ob2, sc, lab, boxes);
  k_keys<<<(BATCH * NSORT + 255) / 256, 256, 0, stream>>>(sc, keys);
  k_sort<<<BATCH, 1024, 0, stream>>>(keys);
  k_gather<<<BATCH, 256, 0, stream>>>(keys, boxes, lab, tb, tsc, tlab, obx, area);
  k_iou<<<dim3(252, BATCH), 256, 0, stream>>>(obx, area, sup);
  k_nms<<<BATCH, 32, 0, stream>>>(sup, tsc, tb, tlab, (float*)d_out);
}